// GraphConvolution_62551903699212
// MI455X (gfx1250) — compile-verified
//
#include <hip/hip_runtime.h>

// GraphConvolution for MI455X (gfx1250, wave32).
// Phase 1: hidden = x @ w via V_WMMA_F32_16X16X4_F32 (exact fp32 matrix pipe).
// Phase 2: out initialized to bias, edge messages scatter-added with
//          coalesced b128 gathers + global_atomic_add_f32 (L2-resident).
// Phase 3: in-place ReLU.

typedef __attribute__((ext_vector_type(2))) float v2f;
typedef __attribute__((ext_vector_type(8))) float v8f;

#define IN_DIM  128
#define OUT_DIM 64

// ---------------------------------------------------------------------------
// Phase 1: dense projection. One wave per 16-row node tile; each wave owns a
// 16x64 output tile held in 4 v8f accumulators. K loop: 128 in steps of 4.
//
// A-frag (16x4 f32, ISA 7.12.2): lane L<16 holds rows M=L, K = k..k+1 in
// v[0],v[1]; lanes 16-31 hold K = k+2..k+3. That is one contiguous 8-byte
// load per lane from x[row][k + 2*(lane>>4)].
// B-frag (4x16 f32): N striped across lanes within a VGPR, K split across
// VGPRs / lane halves, mirroring the C/D layout. Each element is a coalesced
// 64B read of a row of w.
// ---------------------------------------------------------------------------
__global__ __launch_bounds__(256) void gcn_project_wmma(
    const float* __restrict__ x, const float* __restrict__ w,
    float* __restrict__ hidden, int n_tiles) {
  const int lane = threadIdx.x & 31;
  const int wave = threadIdx.x >> 5;
  const int tile = blockIdx.x * 8 + wave;
  if (tile >= n_tiles) return;                 // wave-uniform exit, no barriers
  const int half = lane >> 4;                  // 0 = lanes 0-15, 1 = lanes 16-31
  const int l16  = lane & 15;

  const float* xrow = x + (size_t)(tile * 16 + l16) * IN_DIM;

  v8f c0 = {}, c1 = {}, c2 = {}, c3 = {};
  for (int k = 0; k < IN_DIM; k += 4) {
    const int koff = k + half * 2;
    // A fragment: contiguous pair -> single b64 load per lane.
    v2f a;
    a.x = xrow[koff];
    a.y = xrow[koff + 1];
    // B fragments: coalesced 64B reads of w rows (L0/L2 hot, w = 32KB).
    const float* wr0 = w + (size_t)koff * OUT_DIM + l16;
    const float* wr1 = wr0 + OUT_DIM;
    v2f b0, b1, b2, b3;
    b0.x = wr0[0];  b0.y = wr1[0];
    b1.x = wr0[16]; b1.y = wr1[16];
    b2.x = wr0[32]; b2.y = wr1[32];
    b3.x = wr0[48]; b3.y = wr1[48];

    c0 = __builtin_amdgcn_wmma_f32_16x16x4_f32(false, a, false, b0, (short)0, c0, false, false);
    c1 = __builtin_amdgcn_wmma_f32_16x16x4_f32(false, a, false, b1, (short)0, c1, false, false);
    c2 = __builtin_amdgcn_wmma_f32_16x16x4_f32(false, a, false, b2, (short)0, c2, false, false);
    c3 = __builtin_amdgcn_wmma_f32_16x16x4_f32(false, a, false, b3, (short)0, c3, false, false);
  }

  // C/D layout: VGPR i holds M = i + 8*half, N = l16.
#pragma unroll
  for (int i = 0; i < 8; ++i) {
    const size_t row = (size_t)(tile * 16 + i + 8 * half) * OUT_DIM;
    hidden[row + l16]      = c0[i];
    hidden[row + 16 + l16] = c1[i];
    hidden[row + 32 + l16] = c2[i];
    hidden[row + 48 + l16] = c3[i];
  }
}

// ---------------------------------------------------------------------------
// Phase 2a: seed the output accumulator with the bias (folds the "+ b").
// One thread per 4 output floats.
// ---------------------------------------------------------------------------
__global__ __launch_bounds__(256) void gcn_init_bias(
    float* __restrict__ out, const float* __restrict__ b, long n_vec4) {
  const long gid = (long)blockIdx.x * blockDim.x + threadIdx.x;
  if (gid >= n_vec4) return;
  const int j = (int)(gid & 15) * 4;           // column group within OUT_DIM=64
  const float4 bv = *reinterpret_cast<const float4*>(b + j);
  *(reinterpret_cast<float4*>(out) + gid) = bv;
}

// ---------------------------------------------------------------------------
// Phase 2b: edge scatter. 16 lanes cooperate on one edge: each lane gathers a
// float4 of hidden[col] (one global_load_b128, 256B per edge, L2-resident) and
// issues 4 global_atomic_add_f32 into out[row] (coalesced 256B atomic burst).
// ---------------------------------------------------------------------------
__global__ __launch_bounds__(256) void gcn_edge_scatter(
    const int* __restrict__ erow, const int* __restrict__ ecol,
    const float* __restrict__ eval, const float* __restrict__ hidden,
    float* __restrict__ out, int n_edges) {
  const long gid = (long)blockIdx.x * blockDim.x + threadIdx.x;
  const int e = (int)(gid >> 4);
  if (e >= n_edges) return;
  const int j = (int)(gid & 15) * 4;

  const int   c = ecol[e];
  const int   r = erow[e];
  const float v = eval[e];

  const float4 h = *reinterpret_cast<const float4*>(hidden + (size_t)c * OUT_DIM + j);
  float* o = out + (size_t)r * OUT_DIM + j;
  atomicAdd(o + 0, h.x * v);
  atomicAdd(o + 1, h.y * v);
  atomicAdd(o + 2, h.z * v);
  atomicAdd(o + 3, h.w * v);
}

// ---------------------------------------------------------------------------
// Phase 3: in-place ReLU over the accumulator (= d_out).
// ---------------------------------------------------------------------------
__global__ __launch_bounds__(256) void gcn_relu(float* __restrict__ out, long n_vec4) {
  const long gid = (long)blockIdx.x * blockDim.x + threadIdx.x;
  if (gid >= n_vec4) return;
  float4* p = reinterpret_cast<float4*>(out) + gid;
  float4 t = *p;
  t.x = fmaxf(t.x, 0.f);
  t.y = fmaxf(t.y, 0.f);
  t.z = fmaxf(t.z, 0.f);
  t.w = fmaxf(t.w, 0.f);
  *p = t;
}

extern "C" void kernel_launch(void* const* d_in, const int* in_sizes, int n_in,
                              void* d_out, int out_size, void* d_ws, size_t ws_size,
                              hipStream_t stream) {
  const float* x        = (const float*)d_in[0];
  const int*   edge_row = (const int*)  d_in[1];
  const int*   edge_col = (const int*)  d_in[2];
  const float* edge_val = (const float*)d_in[3];
  const float* w        = (const float*)d_in[4];
  const float* b        = (const float*)d_in[5];
  float*       out      = (float*)d_out;

  const int n_nodes = in_sizes[0] / IN_DIM;   // 50000
  const int n_edges = in_sizes[1];            // 800000
  const int n_tiles = (n_nodes + 15) / 16;    // 3125 (exact)

  float* hidden = (float*)d_ws;               // n_nodes * 64 * 4B = 12.8 MB

  // Phase 1: projection (WMMA). 8 waves per block, 1 tile per wave.
  {
    dim3 grid((n_tiles + 7) / 8), block(256);
    gcn_project_wmma<<<grid, block, 0, stream>>>(x, w, hidden, n_tiles);
  }
  // Phase 2a: out = bias (accumulator seed).
  const long n_vec4 = (long)n_nodes * (OUT_DIM / 4);
  {
    dim3 grid((unsigned)((n_vec4 + 255) / 256)), block(256);
    gcn_init_bias<<<grid, block, 0, stream>>>(out, b, n_vec4);
  }
  // Phase 2b: edge gather + atomic scatter-add.
  {
    const long n_thr = (long)n_edges * 16;
    dim3 grid((unsigned)((n_thr + 255) / 256)), block(256);
    gcn_edge_scatter<<<grid, block, 0, stream>>>(edge_row, edge_col, edge_val,
                                                 hidden, out, n_edges);
  }
  // Phase 3: ReLU in place.
  {
    dim3 grid((unsigned)((n_vec4 + 255) / 256)), block(256);
    gcn_relu<<<grid, block, 0, stream>>>(out, n_vec4);
  }
}